// DetectionLoss_16973710754141
// MI455X (gfx1250) — compile-verified
//
#include <hip/hip_runtime.h>

#define B_  16
#define C_  80
#define H_  64
#define W_  64
#define K_  32
#define P_  (B_ * K_)
#define HW_ (H_ * W_)
#define EPS_ 1e-6f

typedef __attribute__((ext_vector_type(2))) float v2f;
typedef __attribute__((ext_vector_type(8))) float v8f;
typedef __attribute__((ext_vector_type(4))) int   v4i;

// pointer-to-v4i in explicit address spaces (global = 1, LDS = 3)
typedef __attribute__((address_space(1))) v4i* gv4i_p;
typedef __attribute__((address_space(3))) v4i* lv4i_p;

#if defined(__has_builtin)
#  if __has_builtin(__builtin_amdgcn_global_load_async_to_lds_b128)
#    define HAVE_ASYNC_LDS 1
#  endif
#  if __has_builtin(__builtin_amdgcn_s_wait_asynccnt)
#    define HAVE_WAIT_ASYNC_BUILTIN 1
#  endif
#endif

__device__ __forceinline__ void wait_asynccnt0() {
#if defined(HAVE_WAIT_ASYNC_BUILTIN)
    __builtin_amdgcn_s_wait_asynccnt(0);
#else
    asm volatile("s_wait_asynccnt 0" ::: "memory");
#endif
}

// ---------------------------------------------------------------------------
// Kernel 1: CAM supervision loss, one 256-thread block (8 wave32) per GT box.
// Stages the 64x64 cam channel into LDS — via GLOBAL_LOAD_ASYNC_TO_LDS_B128
// (ASYNCcnt path) when the toolchain exposes it, else coalesced float4 — then
// reduces the 16 (4x4) 16x16 tiles with V_WMMA_F32_16X16X4_F32 against a
// ones B-matrix: D[m,n] += sum_k A[m,k] (exact f32 row-sums, replicated over
// n). Two accumulators: total sum and in-box masked sum (branch-free masking
// keeps EXEC all-ones around the WMMAs).
// ---------------------------------------------------------------------------
__global__ __launch_bounds__(256) void camloss_kernel(
    const float* __restrict__ cam, const float* __restrict__ gt_boxes,
    const int* __restrict__ gt_labels, float* __restrict__ ws_term)
{
    __shared__ float chan_lds[HW_];   // 16 KB channel staging
    __shared__ float wave_sums[16];   // [0..7]=total, [8..15]=masked

    const int p = blockIdx.x;         // box id in [0, 512)
    const int b = p / K_;
    const int label = gt_labels[p];
    const float* __restrict__ chan = cam + (size_t)(b * C_ + label) * HW_;

    const int t = threadIdx.x;
#if defined(HAVE_ASYNC_LDS)
    {
        // per-lane 16B async copies: 256 threads x 16B x 4 rounds = 16 KB,
        // written directly into LDS (no VGPR round-trip), tracked by ASYNCcnt.
        const char* gsrc = (const char*)chan + (size_t)t * 16;
        char* ldst = (char*)chan_lds + (size_t)t * 16;
        #pragma unroll
        for (int it = 0; it < 4; ++it) {
            __builtin_amdgcn_global_load_async_to_lds_b128(
                (gv4i_p)(gsrc + it * 4096),
                (lv4i_p)(ldst + it * 4096),
                0, 0);
        }
        wait_asynccnt0();
    }
#else
    {
        // fallback: coalesced stage, 256 threads x float4 x 4 iters
        const float4* __restrict__ src = (const float4*)chan;
        float4* dst = (float4*)chan_lds;
        #pragma unroll
        for (int it = 0; it < 4; ++it)
            dst[t + 256 * it] = src[t + 256 * it];
    }
#endif

    // box -> integer index ranges (truncation == floor for nonneg coords)
    const float x0 = gt_boxes[p * 4 + 0], y0 = gt_boxes[p * 4 + 1];
    const float x1 = gt_boxes[p * 4 + 2], y1 = gt_boxes[p * 4 + 3];
    const int jmin = max(0,      (int)(x0 * (float)W_));
    const int imin = max(0,      (int)(y0 * (float)H_));
    const int jmax = min(W_ - 1, (int)(x1 * (float)W_));
    const int imax = min(H_ - 1, (int)(y1 * (float)H_));

    __syncthreads();

    const int wave = t >> 5;
    const int lane = t & 31;

    v2f ones; ones.x = 1.0f; ones.y = 1.0f;   // B = ones(4x16): 2 f32 per lane
    v8f acc_tot = {};
    v8f acc_in  = {};

    #pragma unroll
    for (int tt = 0; tt < 2; ++tt) {
        const int tile_id = wave * 2 + tt;        // 16 tiles, 2 per wave
        const int ti = (tile_id >> 2) * 16;       // tile row origin
        const int tj = (tile_id & 3)  * 16;       // tile col origin
        const int row = ti + (lane & 15);         // A: lanes 0-15 M=lane, K=0,1
        const bool rin = (row >= imin) & (row <= imax);
        #pragma unroll
        for (int g = 0; g < 4; ++g) {             // 4 K-chunks per 16x16 tile
            const int col = tj + g * 4 + (lane >> 4) * 2;   // lanes 16-31: K=2,3
            const float a0 = chan_lds[row * W_ + col];
            const float a1 = chan_lds[row * W_ + col + 1];
            const float m0 = (rin & (col     >= jmin) & (col     <= jmax)) ? a0 : 0.0f;
            const float m1 = (rin & (col + 1 >= jmin) & (col + 1 <= jmax)) ? a1 : 0.0f;
            v2f A;  A.x  = a0; A.y  = a1;
            v2f Am; Am.x = m0; Am.y = m1;
            acc_tot = __builtin_amdgcn_wmma_f32_16x16x4_f32(
                false, A,  false, ones, (short)0, acc_tot, false, false);
            acc_in  = __builtin_amdgcn_wmma_f32_16x16x4_f32(
                false, Am, false, ones, (short)0, acc_in,  false, false);
        }
    }

    // D[m,n] = rowsum(m) for every n; lanes 0-15 hold m=0..7, lanes 16-31 m=8..15.
    float s_tot = 0.0f, s_in = 0.0f;
    #pragma unroll
    for (int i = 0; i < 8; ++i) { s_tot += acc_tot[i]; s_in += acc_in[i]; }
    s_tot += __shfl_xor(s_tot, 16, 32);   // combine the two row-halves
    s_in  += __shfl_xor(s_in,  16, 32);

    if (lane == 0) { wave_sums[wave] = s_tot; wave_sums[8 + wave] = s_in; }
    __syncthreads();

    if (t == 0) {
        float tot = 0.0f, sin_ = 0.0f;
        #pragma unroll
        for (int wv = 0; wv < 8; ++wv) { tot += wave_sums[wv]; sin_ += wave_sums[8 + wv]; }
        const float cnt_in  = (float)(max(0, imax - imin + 1) * max(0, jmax - jmin + 1));
        const float cnt_out = (float)HW_ - cnt_in;
        const float cam_in  = sin_ / fmaxf(cnt_in, 1.0f);
        const float cam_out = (tot - sin_) / fmaxf(cnt_out, 1.0f);
        const float term = ((cnt_in  > 0.0f) ? (1.0f - cam_in) : 0.0f)
                         + ((cnt_out > 0.0f) ? cam_out : 0.0f);
        ws_term[p] = term;
    }
}

// ---------------------------------------------------------------------------
// Kernel 2: per-positive L1 + GIoU terms (P = 512 gathers, trivial).
// ---------------------------------------------------------------------------
__global__ __launch_bounds__(256) void boxloss_kernel(
    const float* __restrict__ pred_boxes, const float* __restrict__ gt_boxes,
    const int* __restrict__ pos_b, const int* __restrict__ pos_class,
    const int* __restrict__ pos_i, const int* __restrict__ pos_j,
    const int* __restrict__ pos_gt,
    float* __restrict__ ws_l1, float* __restrict__ ws_giou)
{
    const int p = blockIdx.x * 256 + threadIdx.x;
    if (p >= P_) return;
    const int b = pos_b[p], c = pos_class[p], i = pos_i[p], j = pos_j[p], g = pos_gt[p];
    const float* __restrict__ pb =
        pred_boxes + ((((size_t)b * C_ + c) * H_ + i) * W_ + j) * 4;
    const float* __restrict__ gb = gt_boxes + ((size_t)b * K_ + g) * 4;

    const float p0 = pb[0], p1 = pb[1], p2 = pb[2], p3 = pb[3];
    const float g0 = gb[0], g1 = gb[1], g2 = gb[2], g3 = gb[3];

    const float l1 = (fabsf(p0 - g0) + fabsf(p1 - g1) +
                      fabsf(p2 - g2) + fabsf(p3 - g3)) * 0.25f;

    const float inter = fmaxf(fminf(p2, g2) - fmaxf(p0, g0), 0.0f) *
                        fmaxf(fminf(p3, g3) - fmaxf(p1, g1), 0.0f);
    const float a1  = (p2 - p0) * (p3 - p1);
    const float a2  = (g2 - g0) * (g3 - g1);
    const float uni = a1 + a2 - inter;
    const float iou = inter / (uni + EPS_);
    const float enc = (fmaxf(p2, g2) - fminf(p0, g0)) *
                      (fmaxf(p3, g3) - fminf(p1, g1));
    const float giou = iou - (enc - uni) / (enc + EPS_);

    ws_l1[p]   = l1;
    ws_giou[p] = 1.0f - giou;
}

// ---------------------------------------------------------------------------
// Kernel 3: deterministic tree reduction of the three 512-element arrays.
// ---------------------------------------------------------------------------
__global__ __launch_bounds__(256) void finalize_kernel(
    const float* __restrict__ ws_term, const float* __restrict__ ws_l1,
    const float* __restrict__ ws_giou, float* __restrict__ out)
{
    __shared__ float red0[256], red1[256], red2[256];
    const int t = threadIdx.x;
    red0[t] = ws_l1[t]   + ws_l1[t + 256];
    red1[t] = ws_giou[t] + ws_giou[t + 256];
    red2[t] = ws_term[t] + ws_term[t + 256];
    __syncthreads();
    for (int s = 128; s > 0; s >>= 1) {
        if (t < s) {
            red0[t] += red0[t + s];
            red1[t] += red1[t + s];
            red2[t] += red2[t + s];
        }
        __syncthreads();
    }
    if (t == 0) {
        const float inv = 1.0f / (float)P_;
        const float l1 = red0[0] * inv;
        const float gi = red1[0] * inv;
        const float cm = red2[0] * inv;
        out[0] = l1;
        out[1] = gi;
        out[2] = cm;
        out[3] = 1.0f * l1 + 2.0f * gi + 0.5f * cm;
    }
}

// ---------------------------------------------------------------------------
extern "C" void kernel_launch(void* const* d_in, const int* in_sizes, int n_in,
                              void* d_out, int out_size, void* d_ws, size_t ws_size,
                              hipStream_t stream)
{
    (void)in_sizes; (void)n_in; (void)out_size; (void)ws_size;
    const float* cam        = (const float*)d_in[0];
    const float* pred_boxes = (const float*)d_in[1];
    const float* gt_boxes   = (const float*)d_in[2];
    const int*   gt_labels  = (const int*)d_in[3];
    const int*   pos_b      = (const int*)d_in[4];
    const int*   pos_class  = (const int*)d_in[5];
    const int*   pos_i      = (const int*)d_in[6];
    const int*   pos_j      = (const int*)d_in[7];
    const int*   pos_gt     = (const int*)d_in[8];

    float* ws_term = (float*)d_ws;          // [512]
    float* ws_l1   = ws_term + P_;          // [512]
    float* ws_giou = ws_l1   + P_;          // [512]
    float* out     = (float*)d_out;         // [4]

    camloss_kernel<<<P_, 256, 0, stream>>>(cam, gt_boxes, gt_labels, ws_term);
    boxloss_kernel<<<(P_ + 255) / 256, 256, 0, stream>>>(
        pred_boxes, gt_boxes, pos_b, pos_class, pos_i, pos_j, pos_gt,
        ws_l1, ws_giou);
    finalize_kernel<<<1, 256, 0, stream>>>(ws_term, ws_l1, ws_giou, out);
}